// PointsRasterizerPro_15642270892100
// MI455X (gfx1250) — compile-verified
//
#include <hip/hip_runtime.h>
#include <hip/hip_bf16.h>
#include <stdint.h>

// ---------------- problem constants (match reference) ----------------
#define HH      128
#define WW      128
#define KK      8
#define NB      4          // batches
#define NPTS    8192       // points per batch
#define RADIUS  0.02f
#define BIGZ    10000000000.0f
#define EPSW    1e-06f

#define TOTAL_PTS (NB * NPTS)          // 32768
#define OUT_SZ    (NB * HH * WW * KK)  // 524288 per tensor

typedef __attribute__((ext_vector_type(2))) float v2f;
typedef __attribute__((ext_vector_type(8))) float v8f;

// ======================================================================
// Kernel 1: point transform via V_WMMA_F32_16X16X4_F32.
// A (16x4) : 16 points, homogeneous coords (x,y,z,1) along K.
// B (4x16) : cols 0..3 = W  (D cols 0..3 = pv, view_z = col 2)
//            cols 4..7 = W@P (D cols 4..7 = q)
// Result un-striped through a small LDS stage, then NDC divide.
// ======================================================================

__device__ __forceinline__ float bcomb(const float* __restrict__ W,
                                       const float* __restrict__ P,
                                       int k, int col) {
    if (col < 4) return W[k * 4 + col];
    if (col < 8) {
        int j = col - 4;                               // (W @ P)[k][j]
        return W[k*4+0]*P[0*4+j] + W[k*4+1]*P[1*4+j]
             + W[k*4+2]*P[2*4+j] + W[k*4+3]*P[3*4+j];
    }
    return 0.0f;
}

__global__ __launch_bounds__(256)
void transform_wmma_kernel(const float* __restrict__ pts,   // (B,N,3)
                           const float* __restrict__ W,     // 4x4 row-major
                           const float* __restrict__ P,     // 4x4 row-major
                           float4* __restrict__ out)        // (B*N) records
{
    __shared__ float st[8][16][8];   // [wave][row(point)][col] staging

    const int tid  = threadIdx.x;
    const int lane = tid & 31;
    const int wv   = tid >> 5;
    const int gp   = (blockIdx.x * 8 + wv) * 16;  // first point of this wave's group
    const int row  = lane & 15;
    const bool hi  = lane >= 16;

    // ---- A matrix: lanes 0-15 hold K={0,1}=(x,y); lanes 16-31 hold K={2,3}=(z,1)
    const float* pr = pts + (size_t)(gp + row) * 3;
    v2f a;
    a.x = hi ? pr[2] : pr[0];
    a.y = hi ? 1.0f  : pr[1];

    // ---- B matrix: lane n<16 -> B[0][n],B[1][n]; lane n>=16 -> B[2][n-16],B[3][n-16]
    const int col = row;
    const int k0  = hi ? 2 : 0;
    v2f b;
    b.x = bcomb(W, P, k0,     col);
    b.y = bcomb(W, P, k0 + 1, col);

    v8f c = {};
    // (neg_a, A, neg_b, B, c_mod, C, reuse_a, reuse_b) -> v_wmma_f32_16x16x4_f32
    v8f d = __builtin_amdgcn_wmma_f32_16x16x4_f32(
        false, a, false, b, (short)0, c, false, false);

    // ---- un-stripe: D vgpr r holds (M=r, N=lane) lanes 0-15 / (M=r+8) lanes 16-31
    if (col < 8) {
        const int rbase = hi ? 8 : 0;
#pragma unroll
        for (int r = 0; r < 8; ++r)
            st[wv][rbase + r][col] = d[r];
    }
    __syncthreads();

    if (lane < 16) {
        const float vz = st[wv][lane][2];   // pv[2] = view_z
        const float q0 = st[wv][lane][4];
        const float q1 = st[wv][lane][5];
        const float q3 = st[wv][lane][7];   // w
        const float sgn = (q3 >= 0.0f) ? 1.0f : -1.0f;
        const float den = sgn * fmaxf(fabsf(q3), EPSW);
        float4 o;
        o.x = q0 / den;
        o.y = q1 / den;
        o.z = vz;
        o.w = 0.0f;
        out[gp + lane] = o;
    }
}

// ======================================================================
// Kernel 2: tiled rasterizer. One 256-thread workgroup per 16x16 pixel
// tile; 8 passes of 1024-point chunks, each staged into LDS with
// global_load_async_to_lds_b128, culled against the tile bbox, compacted
// in LDS, then scanned by every pixel thread with a sorted top-8 insert.
// ======================================================================

#define CHUNKP 1024                      // points per staged chunk
#define NPASS  (NPTS / CHUNKP)           // 8

__global__ __launch_bounds__(256)
void raster_tile_kernel(const float4* __restrict__ tp,   // transformed pts (B*N)
                        int*   __restrict__ idx_out0,
                        float* __restrict__ zbuf_out,
                        int*   __restrict__ idx_out1,
                        float* __restrict__ d2_out)
{
    __shared__ float4 stage[CHUNKP];     // raw chunk (async-loaded)  16 KB
    __shared__ float4 comp[CHUNKP];      // culled survivors          16 KB
    __shared__ int    cnt;

    const int tid   = threadIdx.x;
    const int tx    = tid & 15;
    const int ty    = tid >> 4;
    const int blk   = blockIdx.x;
    const int tilew = blk & 7;           // W/16 = 8
    const int tileh = (blk >> 3) & 7;    // H/16 = 8
    const int bb    = blk >> 6;          // batch
    const int w     = tilew * 16 + tx;
    const int h     = tileh * 16 + ty;

    const float px = 1.0f - (2.0f * (float)w + 1.0f) * (1.0f / (float)WW);
    const float py = 1.0f - (2.0f * (float)h + 1.0f) * (1.0f / (float)HH);

    // tile bbox in NDC (px/py decrease with w/h), expanded by RADIUS
    const float xmax = 1.0f - (2.0f * (float)(tilew * 16)      + 1.0f) / (float)WW + RADIUS;
    const float xmin = 1.0f - (2.0f * (float)(tilew * 16 + 15) + 1.0f) / (float)WW - RADIUS;
    const float ymax = 1.0f - (2.0f * (float)(tileh * 16)      + 1.0f) / (float)HH + RADIUS;
    const float ymin = 1.0f - (2.0f * (float)(tileh * 16 + 15) + 1.0f) / (float)HH - RADIUS;

    float zk[KK]; int ik[KK]; float dk[KK];
#pragma unroll
    for (int k = 0; k < KK; ++k) { zk[k] = BIGZ; ik[k] = -1; dk[k] = 0.0f; }

    const float4* src_b = tp + (size_t)bb * NPTS;
    const uint32_t lds_stage = (uint32_t)(uintptr_t)(&stage[0]);
    const float r2 = RADIUS * RADIUS;

    for (int pass = 0; pass < NPASS; ++pass) {
        if (tid == 0) cnt = 0;

        // ---- async stage: 1024 float4 records -> LDS (4 x B128 per lane)
        const float4* src = src_b + pass * CHUNKP;
#pragma unroll
        for (int i = 0; i < CHUNKP / 256; ++i) {
            const int j = i * 256 + tid;
            const uint32_t loff = lds_stage + (uint32_t)j * 16u;
            const void* ga = (const void*)(src + j);
            asm volatile("global_load_async_to_lds_b128 %0, %1, off"
                         :: "v"(loff), "v"(ga) : "memory");
        }
        // prefetch next chunk toward L2 while we wait
        if (pass + 1 < NPASS)
            __builtin_prefetch((const void*)(src + CHUNKP + tid), 0, 1);
        asm volatile("s_wait_asynccnt 0" ::: "memory");
        __syncthreads();

        // ---- cull to tile bbox, compact into LDS list
#pragma unroll
        for (int i = 0; i < CHUNKP / 256; ++i) {
            const int j = i * 256 + tid;
            float4 p = stage[j];
            if (p.z > 0.0f &&
                p.x >= xmin && p.x <= xmax &&
                p.y >= ymin && p.y <= ymax) {
                const int pos = atomicAdd(&cnt, 1);
                p.w = __int_as_float(pass * CHUNKP + j);   // global point index
                comp[pos] = p;
            }
        }
        __syncthreads();

        // ---- per-pixel scan of survivors (LDS broadcast reads)
        int n = cnt; if (n > CHUNKP) n = CHUNKP;
        for (int m = 0; m < n; ++m) {
            const float4 cd = comp[m];
            const float dx = cd.x - px;
            const float dy = cd.y - py;
            const float d2 = dx * dx + dy * dy;
            const float zc = cd.z;
            if (d2 <= r2 && zc < zk[KK - 1]) {
                const int   ic = __float_as_int(cd.w);
#pragma unroll
                for (int k = KK - 1; k >= 0; --k) {
                    if (zc < zk[k]) {
                        if (k < KK - 1) { zk[k+1] = zk[k]; ik[k+1] = ik[k]; dk[k+1] = dk[k]; }
                        zk[k] = zc; ik[k] = ic; dk[k] = d2;
                    }
                }
            }
        }
        __syncthreads();   // before cnt reset / stage overwrite next pass
    }

    // ---- write outputs (idx, zbuf, idx, dists2), masking invalid slots
    const int base = (((bb * HH) + h) * WW + w) * KK;
#pragma unroll
    for (int k = 0; k < KK; ++k) {
        const bool valid = zk[k] < 0.5f * BIGZ;
        const int   io = valid ? ik[k] : -1;
        const float zo = valid ? zk[k] : -1.0f;
        const float dO = valid ? dk[k] : -1.0f;
        idx_out0[base + k] = io;
        zbuf_out[base + k] = zo;
        idx_out1[base + k] = io;
        d2_out  [base + k] = dO;
    }
}

// ======================================================================
// Host launcher
// ======================================================================
extern "C" void kernel_launch(void* const* d_in, const int* in_sizes, int n_in,
                              void* d_out, int out_size, void* d_ws, size_t ws_size,
                              hipStream_t stream) {
    (void)in_sizes; (void)n_in; (void)out_size; (void)ws_size;

    const float* pts = (const float*)d_in[0];   // (4,8192,3) f32
    const float* W   = (const float*)d_in[1];   // (4,4) f32
    const float* P   = (const float*)d_in[2];   // (4,4) f32

    float4* tpts = (float4*)d_ws;               // 32768 * 16 B = 512 KB scratch

    // transform: 32768 pts / (8 waves * 16 pts per block) = 256 blocks
    transform_wmma_kernel<<<256, 256, 0, stream>>>(pts, W, P, tpts);

    int*   idx0 = (int*)d_out;
    float* zb   = (float*)d_out + OUT_SZ;
    int*   idx1 = (int*)d_out + 2 * OUT_SZ;
    float* dd   = (float*)d_out + 3 * OUT_SZ;

    // raster: 4 batches * 8*8 tiles = 256 blocks of 256 threads
    raster_tile_kernel<<<256, 256, 0, stream>>>(tpts, idx0, zb, idx1, dd);
}